// Group_84567906058595
// MI455X (gfx1250) — compile-verified
//
#include <hip/hip_runtime.h>

// Problem dims (fixed by the reference)
#define BB 4
#define NN 32768
#define GG 2048
#define KK 32
#define DD 128

// Output layout (floats): neighborhood | center | emb_g
#define NB_OFF 0
#define C_OFF  (BB * GG * KK * 3)          // 786432
#define E_OFF  (C_OFF + BB * GG * 3)       // 811008

// Tiling
#define NCHUNK 1024
#define LSTR   (NCHUNK + 4)                // LDS row stride (pad to dodge bank conflicts)

typedef __attribute__((ext_vector_type(2))) float v2f;
typedef __attribute__((ext_vector_type(8))) float v8f;

__device__ __forceinline__ int lane_id() { return (int)(threadIdx.x & 31u); }

// wave32 xor-shuffles via ds_bpermute (byte index, wraps at 32 lanes on wave32)
__device__ __forceinline__ float shflx_f(float v, int m) {
    int idx = ((lane_id() ^ m) << 2);
    return __int_as_float(__builtin_amdgcn_ds_bpermute(idx, __float_as_int(v)));
}
__device__ __forceinline__ int shflx_i(int v, int m) {
    int idx = ((lane_id() ^ m) << 2);
    return __builtin_amdgcn_ds_bpermute(idx, v);
}

// ---------------------------------------------------------------------------
// Kernel 1: pack points as (x, y, z, x^2+y^2+z^2) — small, L2-resident table
// ---------------------------------------------------------------------------
__global__ void prep_points(const float* __restrict__ xyz, float4* __restrict__ P) {
    int t = blockIdx.x * blockDim.x + threadIdx.x;   // over B*N
    if (t >= BB * NN) return;
    float x = xyz[3 * t + 0];
    float y = xyz[3 * t + 1];
    float z = xyz[3 * t + 2];
    P[t] = make_float4(x, y, z, x * x + y * y + z * z);
}

// ---------------------------------------------------------------------------
// Kernel 2: gather centers; emit `center` output and WMMA A-rows (-2c, 1)
// ---------------------------------------------------------------------------
__global__ void prep_centers(const float* __restrict__ xyz,
                             const int* __restrict__ center_idx,
                             float4* __restrict__ A4,
                             float* __restrict__ out_center) {
    int t = blockIdx.x * blockDim.x + threadIdx.x;   // over B*G
    if (t >= BB * GG) return;
    int b = t / GG;
    int g = t - b * GG;
    int c = center_idx[g];
    const float* p = xyz + ((size_t)b * NN + (size_t)c) * 3;
    float x = p[0], y = p[1], z = p[2];
    out_center[3 * t + 0] = x;
    out_center[3 * t + 1] = y;
    out_center[3 * t + 2] = z;
    A4[t] = make_float4(-2.0f * x, -2.0f * y, -2.0f * z, 1.0f);
}

// ---------------------------------------------------------------------------
// Kernel 3: fused WMMA distance tiles + per-row wave top-K + gathers
// grid = (G/16, B), block = 256 (8 waves). Wave w selects rows 2w, 2w+1.
// ---------------------------------------------------------------------------
__global__ void __launch_bounds__(256)
group_topk(const float4* __restrict__ P,
           const float4* __restrict__ A4,
           const float*  __restrict__ emb,
           float* __restrict__ out) {
    __shared__ float lds_dist[16 * LSTR];

    const int b     = blockIdx.y;
    const int gbase = blockIdx.x * 16;
    const int lane  = lane_id();
    const int w     = (int)(threadIdx.x >> 5);       // wave id 0..7
    const int lmod  = lane & 15;
    const bool lo   = (lane < 16);

    // A fragment for this 16-center tile (ISA 16x4 f32 A layout):
    // v0 = K=0 (lanes 0-15) / K=2 (lanes 16-31); v1 = K=1 / K=3. M = lane%16.
    float4 av = A4[(size_t)b * GG + (gbase + lmod)];
    v2f afrag;
    afrag.x = lo ? av.x : av.z;
    afrag.y = lo ? av.y : av.w;

    // Running top-K state: lane l of a wave holds one candidate of the row's top-32
    float bd[2];  int bi[2];  float wst[2];
    bd[0] = bd[1] = 3.0e38f;
    bi[0] = bi[1] = 0;
    wst[0] = wst[1] = 3.0e38f;

    for (int nbase = 0; nbase < NN; nbase += NCHUNK) {
        // ---- compute phase: 64 WMMA 16x16 tiles, 8 per wave ----
        #pragma unroll
        for (int tt = 0; tt < 8; ++tt) {
            int t = w * 8 + tt;
            int n = nbase + t * 16 + lmod;
            float4 pv = P[(size_t)b * NN + n];
            v2f bfrag;                                  // 4x16 f32 B layout (mirror of A)
            bfrag.x = lo ? pv.x : pv.z;
            bfrag.y = lo ? pv.y : pv.w;
            v8f cz = {};
            v8f d = __builtin_amdgcn_wmma_f32_16x16x4_f32(
                        false, afrag, false, bfrag, (short)0, cz, false, false);
            int col = t * 16 + lmod;
            int rofs = lo ? 0 : 8;                      // D layout: VGPR r -> M=r / M=r+8
            #pragma unroll
            for (int r = 0; r < 8; ++r)
                lds_dist[(r + rofs) * LSTR + col] = d[r];
        }
        __syncthreads();

        // ---- selection phase: wave w owns rows 2w, 2w+1 ----
        #pragma unroll
        for (int rr = 0; rr < 2; ++rr) {
            int row = 2 * w + rr;
            for (int j = 0; j < NCHUNK / 32; ++j) {
                float cd = lds_dist[row * LSTR + j * 32 + lane];
                if (nbase == 0 && j == 0) {
                    // seed the set with the first 32 candidates
                    bd[rr] = cd;
                    bi[rr] = lane;
                    float x = cd;
                    #pragma unroll
                    for (int off = 16; off > 0; off >>= 1)
                        x = fmaxf(x, shflx_f(x, off));
                    wst[rr] = x;
                } else {
                    unsigned mask = __builtin_amdgcn_ballot_w32(cd < wst[rr]);
                    while (mask) {
                        int s = __builtin_ctz(mask);
                        mask &= mask - 1;
                        float dn = __int_as_float(
                            __builtin_amdgcn_readlane(__float_as_int(cd), s));
                        int in = nbase + j * 32 + s;
                        if (dn < wst[rr]) {               // uniform branch
                            // argmax over the 32-entry set (butterfly, det. tie-break)
                            float md = bd[rr]; int ml = lane;
                            #pragma unroll
                            for (int off = 16; off > 0; off >>= 1) {
                                float od = shflx_f(md, off);
                                int   ol = shflx_i(ml, off);
                                if (od > md || (od == md && ol < ml)) { md = od; ml = ol; }
                            }
                            if (lane == ml) { bd[rr] = dn; bi[rr] = in; }
                            // refresh threshold = new max of set
                            float x = bd[rr];
                            #pragma unroll
                            for (int off = 16; off > 0; off >>= 1)
                                x = fmaxf(x, shflx_f(x, off));
                            wst[rr] = x;
                        }
                    }
                }
            }
        }
        __syncthreads();
    }

    // ---- finalize: sort each row's 32 candidates ascending, then gather ----
    const float4* embv = (const float4*)emb;
    float4*       outE = (float4*)(out + E_OFF);

    #pragma unroll
    for (int rr = 0; rr < 2; ++rr) {
        float d = bd[rr];
        int   i = bi[rr];
        // 32-lane bitonic sort ascending by (dist, idx)
        #pragma unroll
        for (int k = 2; k <= 32; k <<= 1) {
            #pragma unroll
            for (int j = k >> 1; j > 0; j >>= 1) {
                float od = shflx_f(d, j);
                int   oi = shflx_i(i, j);
                bool up    = ((lane & k) == 0);
                bool lower = ((lane & j) == 0);
                bool less  = (od < d) || (od == d && oi < i);
                bool take  = (lower == up) ? less : !less;
                if (take) { d = od; i = oi; }
            }
        }

        int g = gbase + 2 * w + rr;
        size_t go = (size_t)b * GG + g;

        // neighborhood: lane l writes its own neighbor, centered
        float4 pv = P[(size_t)b * NN + i];
        float4 ar = A4[go];
        float cx = -0.5f * ar.x, cy = -0.5f * ar.y, cz2 = -0.5f * ar.z;
        size_t nb = (go * KK + lane) * 3 + NB_OFF;
        out[nb + 0] = pv.x - cx;
        out[nb + 1] = pv.y - cy;
        out[nb + 2] = pv.z - cz2;

        // embeddings: wave copies one 128-float row per neighbor (32 lanes x f32x4)
        for (int l = 0; l < KK; ++l) {
            int nl = __builtin_amdgcn_readlane(i, l);
            float4 v = embv[((size_t)b * NN + nl) * (DD / 4) + lane];
            outE[(go * KK + l) * (DD / 4) + lane] = v;
        }
    }
}

// ---------------------------------------------------------------------------
extern "C" void kernel_launch(void* const* d_in, const int* in_sizes, int n_in,
                              void* d_out, int out_size, void* d_ws, size_t ws_size,
                              hipStream_t stream) {
    (void)in_sizes; (void)n_in; (void)out_size; (void)ws_size;
    const float* xyz = (const float*)d_in[0];
    const float* emb = (const float*)d_in[1];
    const int*   cix = (const int*)d_in[2];
    float* out = (float*)d_out;

    float4* P  = (float4*)d_ws;                                   // B*N float4 = 2 MB
    float4* A4 = (float4*)((char*)d_ws + (size_t)BB * NN * 16);   // B*G float4 = 128 KB

    prep_points <<<(BB * NN) / 256, 256, 0, stream>>>(xyz, P);
    prep_centers<<<(BB * GG) / 256, 256, 0, stream>>>(xyz, cix, A4, out + C_OFF);
    group_topk  <<<dim3(GG / 16, BB), 256, 0, stream>>>(P, A4, emb, out);
}